// BinsChamferLoss_5497558138913
// MI455X (gfx1250) — compile-verified
//
#include <hip/hip_runtime.h>

// ---------------------------------------------------------------------------
// BinsChamferLoss on MI455X (gfx1250).
// Distance tiles computed on the matrix unit via the rank-3 identity
//   (c - p)^2 = c^2*1 + 1*p^2 + (-2c)*p
// using V_WMMA_F32_16X16X4_F32 (16 bins x 16 points per instruction, f32).
// ---------------------------------------------------------------------------

typedef __attribute__((ext_vector_type(2))) float v2f;
typedef __attribute__((ext_vector_type(8))) float v8f;

constexpr int   BATCH      = 4;
constexpr int   NB         = 256;                 // bins per batch
constexpr int   V          = 352 * 448;           // points per batch = 157696
constexpr int   PTS_PER_WG = 2048;                // 8 waves * 16 tiles * 16 pts
constexpr int   WGS_PER_B  = V / PTS_PER_WG;      // 77 (exact)
constexpr int   NWG        = BATCH * WGS_PER_B;   // 308
constexpr float BIGF       = 1e30f;
constexpr float FAR_COORD  = 1e15f;               // invalid points -> dist ~1e30

static_assert(V % PTS_PER_WG == 0, "point count must tile evenly");

// ---------------------------- workspace layout -----------------------------
// [0, 1024)            : gMinB   (uint bit-pattern of f32 min, per batch*bin)
// [1024, 1024+308)     : sumA_part (float, one slot per workgroup)
// [1332, 1336)         : nvalid  (int, per batch)

__global__ __launch_bounds__(256) void init_ws_kernel(unsigned* gMinB,
                                                      int* nvalid) {
  int t = threadIdx.x;
  for (int i = t; i < BATCH * NB; i += 256) gMinB[i] = __float_as_uint(BIGF);
  if (t < BATCH) nvalid[t] = 0;
}

__global__ __launch_bounds__(256) void chamfer_main_kernel(
    const float* __restrict__ bins,   // [BATCH, NB]
    const float* __restrict__ gt,     // [BATCH, V]
    unsigned* __restrict__ gMinB,     // [BATCH*NB] uint-ordered f32 mins
    float* __restrict__ sumA_part,    // [NWG]
    int* __restrict__ nvalid) {       // [BATCH]
  const int tid   = threadIdx.x;
  const int lane  = tid & 31;
  const int wv    = tid >> 5;
  const int b     = blockIdx.x / WGS_PER_B;
  const int chunk = blockIdx.x % WGS_PER_B;
  const int n     = lane & 15;          // point / bin sub-index within half
  const bool loHalf = (lane < 16);

  __shared__ unsigned ldsMinB[NB];
  __shared__ float    ldsSumA[8];
  __shared__ int      ldsCnt[8];

  for (int i = tid; i < NB; i += 256) ldsMinB[i] = __float_as_uint(BIGF);
  __syncthreads();

  // ---- build the 16 A matrices (16 bins each) in registers ----
  // A row m = [c^2, 1, -2c, 0]; lanes 0-15 carry K=0/1, lanes 16-31 K=2/3.
  const float* binsB = bins + b * NB;
  float a0[16], a1[16];
#pragma unroll
  for (int g = 0; g < 16; ++g) {
    float cm = binsB[g * 16 + n];
    a0[g] = loHalf ? cm * cm : -2.0f * cm;
    a1[g] = loHalf ? 1.0f : 0.0f;
  }

  v8f minAcc[16];
#pragma unroll
  for (int g = 0; g < 16; ++g)
#pragma unroll
    for (int v = 0; v < 8; ++v) minAcc[g][v] = BIGF;

  float sumA = 0.0f;
  int   cnt  = 0;

  const float* pBase = gt + b * V + chunk * PTS_PER_WG + wv * 256;

  for (int t = 0; t < 16; ++t) {
    const float* pPtr = pBase + t * 16;
    if (t + 1 < 16) __builtin_prefetch(pPtr + 16, 0, 3);  // global_prefetch_b8

    float p     = pPtr[n];                 // both halves load same 16 points
    bool  valid = (p >= 0.001f);
    float pe    = valid ? p : FAR_COORD;

    // B col n = [1, p^2, p, 0]^T; lanes 0-15 carry K=0/1, lanes 16-31 K=2/3.
    v2f bm;
    bm[0] = loHalf ? 1.0f : pe;
    bm[1] = loHalf ? pe * pe : 0.0f;

    float dmin1 = BIGF;                    // per-lane column (point) min
#pragma unroll
    for (int g = 0; g < 16; ++g) {
      v2f am; am[0] = a0[g]; am[1] = a1[g];
      v8f cz = {0.f, 0.f, 0.f, 0.f, 0.f, 0.f, 0.f, 0.f};
      v8f d = __builtin_amdgcn_wmma_f32_16x16x4_f32(
          false, am, false, bm, (short)0, cz, false, false);

      // direction 2: running elementwise min across point tiles
#pragma unroll
      for (int v = 0; v < 8; ++v) minAcc[g][v] = fminf(minAcc[g][v], d[v]);

      // direction 1: min over the 8 M-rows held by this lane
      float m01 = fminf(d[0], d[1]);
      float m23 = fminf(d[2], d[3]);
      float m45 = fminf(d[4], d[5]);
      float m67 = fminf(d[6], d[7]);
      dmin1 = fminf(dmin1, fminf(fminf(m01, m23), fminf(m45, m67)));
    }
    // combine halves: lanes<16 had rows m%16<8, lanes>=16 rows m%16>=8
    float dfull = fminf(dmin1, __shfl_xor(dmin1, 16));
    bool take = loHalf && valid;
    sumA += take ? dfull : 0.0f;
    cnt  += take ? 1 : 0;
  }

  // ---- direction 2: reduce each bin row over its 16 lanes, merge in LDS ----
#pragma unroll
  for (int g = 0; g < 16; ++g) {
#pragma unroll
    for (int v = 0; v < 8; ++v) {
      float m = minAcc[g][v];
      m = fminf(m, __shfl_xor(m, 1));
      m = fminf(m, __shfl_xor(m, 2));
      m = fminf(m, __shfl_xor(m, 4));
      m = fminf(m, __shfl_xor(m, 8));   // all lanes of each half hold row-min
      if ((lane & 15) == 0) {
        int row = g * 16 + v + ((lane >> 4) << 3);  // lane0 -> v, lane16 -> v+8
        atomicMin(&ldsMinB[row], __float_as_uint(m));
      }
    }
  }

  // ---- direction 1: wave reduction (fixed tree -> deterministic) ----
  float s = sumA;
  s += __shfl_xor(s, 16);
  s += __shfl_xor(s, 8);
  s += __shfl_xor(s, 4);
  s += __shfl_xor(s, 2);
  s += __shfl_xor(s, 1);
  int c = cnt;
  c += __shfl_xor(c, 16);
  c += __shfl_xor(c, 8);
  c += __shfl_xor(c, 4);
  c += __shfl_xor(c, 2);
  c += __shfl_xor(c, 1);
  if (lane == 0) { ldsSumA[wv] = s; ldsCnt[wv] = c; }
  __syncthreads();

  // merge per-bin mins into global (uint-ordered f32; min is order-invariant)
  atomicMin(&gMinB[b * NB + tid], ldsMinB[tid]);

  if (tid == 0) {
    float tot = 0.0f; int ct = 0;
    for (int i = 0; i < 8; ++i) { tot += ldsSumA[i]; ct += ldsCnt[i]; }
    sumA_part[blockIdx.x] = tot;          // deterministic slot per WG
    atomicAdd(&nvalid[b], ct);            // integer -> deterministic
  }
}

__global__ __launch_bounds__(256) void chamfer_finalize_kernel(
    const float* __restrict__ bins,
    const unsigned* __restrict__ gMinB,
    const float* __restrict__ sumA_part,
    const int* __restrict__ nvalid,
    float* __restrict__ out) {
  const int tid = threadIdx.x;
  __shared__ float red[256];

  int n0 = nvalid[0], n1 = nvalid[1], n2 = nvalid[2], n3 = nvalid[3];
  int Lmax = max(max(n0, n1), max(n2, n3));
  int nArr[4] = {n0, n1, n2, n3};

  float acc = 0.0f;
  for (int b = 0; b < BATCH; ++b) {
    float npad = (float)(Lmax - nArr[b]);
    bool  hasPad = npad > 0.0f;

    float cv = bins[b * NB + tid];
    float c2 = cv * cv;
    float mb = __uint_as_float(gMinB[b * NB + tid]);
    float valB = hasPad ? fminf(mb, c2) : mb;

    // sumB: fixed-tree sum over 256 bins
    red[tid] = valB;
    __syncthreads();
    for (int s = 128; s > 0; s >>= 1) {
      if (tid < s) red[tid] += red[tid + s];
      __syncthreads();
    }
    float sumB = red[0];
    __syncthreads();

    // min over c^2
    red[tid] = c2;
    __syncthreads();
    for (int s = 128; s > 0; s >>= 1) {
      if (tid < s) red[tid] = fminf(red[tid], red[tid + s]);
      __syncthreads();
    }
    float minC2 = red[0];
    __syncthreads();

    // sumA: fixed-tree sum over the 77 per-WG partials
    red[tid] = (tid < WGS_PER_B) ? sumA_part[b * WGS_PER_B + tid] : 0.0f;
    __syncthreads();
    for (int s = 128; s > 0; s >>= 1) {
      if (tid < s) red[tid] += red[tid + s];
      __syncthreads();
    }
    float sumA = red[0] + npad * minC2;
    __syncthreads();

    if (tid == 0) acc += sumA + sumB;
  }
  if (tid == 0) out[0] = acc * (1.0f / (float)BATCH);
}

extern "C" void kernel_launch(void* const* d_in, const int* in_sizes, int n_in,
                              void* d_out, int out_size, void* d_ws, size_t ws_size,
                              hipStream_t stream) {
  (void)in_sizes; (void)n_in; (void)out_size; (void)ws_size;
  const float* bins = (const float*)d_in[0];   // [4,256,1] f32
  const float* gt   = (const float*)d_in[1];   // [4,1,352,448] f32
  float* out = (float*)d_out;

  unsigned* gMinB     = (unsigned*)d_ws;                    // 1024 u32
  float*    sumA_part = (float*)(gMinB + BATCH * NB);       // 308 f32
  int*      nvalid    = (int*)(sumA_part + NWG);            // 4 i32

  init_ws_kernel<<<1, 256, 0, stream>>>(gMinB, nvalid);
  chamfer_main_kernel<<<NWG, 256, 0, stream>>>(bins, gt, gMinB, sumA_part,
                                               nvalid);
  chamfer_finalize_kernel<<<1, 256, 0, stream>>>(bins, gMinB, sumA_part,
                                                 nvalid, out);
}